// EfficientCorrelation_28080496181768
// MI455X (gfx1250) — compile-verified
//
#include <hip/hip_runtime.h>

typedef __attribute__((ext_vector_type(2))) float v2f;
typedef __attribute__((ext_vector_type(8))) float v8f;

namespace {
constexpr int kB = 8, kC = 128, kH = 96, kW = 160;
constexpr int kPad = 4, kWin = 9, kOff = 81;   // 9x9 = 81 displacements
constexpr int kTileW = 16;                     // output pixels per block (M)
constexpr int kXCols = 24;                     // staged x2 columns: w0-4 .. w0+19
constexpr int kCHalf = 64;                     // channels staged per pass
constexpr int kCPad  = 68;                     // LDS channel stride (bank-friendly, 8B aligned)
constexpr int kWaves = 9;                      // one wave per dy
constexpr int kThreads = kWaves * 32;
}

__global__ __launch_bounds__(kThreads, 2)
void corr81_wmma_f32(const float* __restrict__ x1,
                     const float* __restrict__ x2,
                     float* __restrict__ out) {
  __shared__ float sA[kTileW * kCPad];          // x1 transposed: [pixel m][channel c]
  __shared__ float sB[kWin * kXCols * kCPad];   // x2 transposed: [dy][col j][channel c]

  const int w0  = blockIdx.x * kTileW;
  const int h   = blockIdx.y;
  const int b   = blockIdx.z;
  const int tid = threadIdx.x;
  const int lane = tid & 31;
  const int wave = tid >> 5;                    // dy index 0..8
  const int n    = lane & 15;                   // N (t=0 col / A pixel row)
  const int hi   = lane >> 4;                   // K-pair select in fragments

  v8f acc0 = {};                                // N-tile j in [0,16)
  v8f acc1 = {};                                // N-tile j in [8,24)

  const float* aPtr  = &sA[n * kCPad + 2 * hi];
  const float* b0Ptr = &sB[(wave * kXCols + n) * kCPad + 2 * hi];
  const float* b1Ptr = &sB[(wave * kXCols + n + 8) * kCPad + 2 * hi];

  for (int half = 0; half < 2; ++half) {
    const int c0 = half * kCHalf;

    // ---- stage x1 tile (transposed) ----
    for (int i = tid; i < kCHalf * kTileW; i += kThreads) {
      int c = i >> 4, m = i & 15;
      sA[m * kCPad + c] = x1[((b * kC + (c0 + c)) * kH + h) * kW + (w0 + m)];
    }
    // ---- stage x2 tile: 9 replicate-clamped rows x 24 clamped cols, transposed ----
    for (int i = tid; i < kWin * kCHalf * kXCols; i += kThreads) {
      int dy = i / (kCHalf * kXCols);
      int r  = i - dy * (kCHalf * kXCols);
      int c  = r / kXCols;
      int j  = r - c * kXCols;
      int gh = h + dy - kPad;  gh = gh < 0 ? 0 : (gh > kH - 1 ? kH - 1 : gh);
      int gw = w0 + j - kPad;  gw = gw < 0 ? 0 : (gw > kW - 1 ? kW - 1 : gw);
      sB[(dy * kXCols + j) * kCPad + c] = x2[((b * kC + (c0 + c)) * kH + gh) * kW + gw];
    }
    __syncthreads();

    // ---- K loop: 16 steps of 16x16x4 fp32 WMMA per half ----
    #pragma unroll 4
    for (int k = 0; k < kCHalf / 4; ++k) {
      v2f a   = *(const v2f*)(aPtr  + 4 * k);   // A 16x4: lane M=n, K pair via hi
      v2f bv0 = *(const v2f*)(b0Ptr + 4 * k);   // B 4x16: lane N=n
      v2f bv1 = *(const v2f*)(b1Ptr + 4 * k);   // B 4x16: lane N=n (cols +8)
      acc0 = __builtin_amdgcn_wmma_f32_16x16x4_f32(
          false, a, false, bv0, (short)0, acc0, false, false);
      acc1 = __builtin_amdgcn_wmma_f32_16x16x4_f32(
          false, a, false, bv1, (short)0, acc1, false, false);
    }
    __syncthreads();   // accumulators live; safe to restage LDS
  }

  // ---- band extraction: element (v,lane) -> M = v+8*hi, N = n; j = N + 8*t ----
  constexpr float kScale = 0.125f;  // 1 / num_chunks (ceil(128/16) = 8)
  #pragma unroll
  for (int v = 0; v < 8; ++v) {
    const int M = v + 8 * hi;
    const int d0 = n - M;                       // t=0: j = n, dx+4 = j - M
    if (d0 >= 0 && d0 <= 8)
      out[((b * kOff + (wave * kWin + d0)) * kH + h) * kW + (w0 + M)] = acc0[v] * kScale;
    const int j1 = n + 8;                       // t=1: j = n + 8, handles j >= 16 only
    const int d1 = j1 - M;
    if (j1 >= 16 && d1 >= 0 && d1 <= 8)
      out[((b * kOff + (wave * kWin + d1)) * kH + h) * kW + (w0 + M)] = acc1[v] * kScale;
  }
}

extern "C" void kernel_launch(void* const* d_in, const int* in_sizes, int n_in,
                              void* d_out, int out_size, void* d_ws, size_t ws_size,
                              hipStream_t stream) {
  (void)in_sizes; (void)n_in; (void)d_ws; (void)ws_size; (void)out_size;
  const float* x1 = (const float*)d_in[0];
  const float* x2 = (const float*)d_in[1];
  float* out = (float*)d_out;
  dim3 grid(kW / kTileW, kH, kB);   // 10 x 96 x 8 = 7680 workgroups
  dim3 block(kThreads);             // 288 threads = 9 wave32 (one per dy)
  hipLaunchKernelGGL(corr81_wmma_f32, grid, block, 0, stream, x1, x2, out);
}